// Attention_49177375539262
// MI455X (gfx1250) — compile-verified
//
#include <hip/hip_runtime.h>

typedef __attribute__((ext_vector_type(2))) float v2f;
typedef __attribute__((ext_vector_type(8))) float v8f;

#define LEN 256
#define DIM 64
#define NTILE 16   // 16 column tiles of 16
#define NK 16      // K steps of 4 (K=64 total)
#define THREADS 512

#if defined(__has_builtin)
#  if __has_builtin(__builtin_amdgcn_sched_barrier)
#    define SCHED_FENCE() __builtin_amdgcn_sched_barrier(0)
#  endif
#endif
#ifndef SCHED_FENCE
#  define SCHED_FENCE()
#endif

__global__ __launch_bounds__(THREADS)
void fused_sigma_attn(const float* __restrict__ q, const float* __restrict__ k,
                      const float* __restrict__ Wq, const float* __restrict__ bq,
                      const float* __restrict__ Wk, const float* __restrict__ bk,
                      const float* __restrict__ W1, const float* __restrict__ W2a,
                      const float* __restrict__ W2b, float* __restrict__ out)
{
    extern __shared__ float smem[];
    float* sQ      = smem;                 // LEN*DIM = 16384 floats
    float* sK      = sQ + LEN * DIM;       // 16384 floats
    float* sSq     = sK + LEN * DIM;       // 256
    float* sSk     = sSq + LEN;            // 256
    float* sTheta  = sSk + LEN;            // 256
    float* sRowMax = sTheta + LEN;         // 256
    float* sRowSum = sRowMax + LEN;        // 256
    float* sRed    = sRowSum + LEN;        // 256

    const int g   = blockIdx.x;            // group id 0..511
    const int tid = threadIdx.x;
    const size_t base = (size_t)g * LEN * DIM;

    // ---- stage q,k tiles into LDS (coalesced float4 copies) ----
    {
        const float4* q4 = (const float4*)(q + base);
        const float4* k4 = (const float4*)(k + base);
        float4* sQ4 = (float4*)sQ;
        float4* sK4 = (float4*)sK;
        const int n4 = LEN * DIM / 4;      // 4096
        for (int i = tid; i < n4; i += THREADS) {
            sQ4[i] = q4[i];
            sK4[i] = k4[i];
        }
    }
    __syncthreads();

    // ---- scalar projections: Sq[i] = q[i]·Wq + bq, Sk[j] = k[j]·Wk + bk ----
    if (tid < LEN) {
        float acc = 0.f;
        #pragma unroll 8
        for (int d = 0; d < DIM; ++d) acc += sQ[tid * DIM + d] * Wq[d];
        sSq[tid] = acc + bq[0];
    } else {
        const int r = tid - LEN;
        float acc = 0.f;
        #pragma unroll 8
        for (int d = 0; d < DIM; ++d) acc += sK[r * DIM + d] * Wk[d];
        sSk[r] = acc + bk[0];
    }
    __syncthreads();

    const int wave   = tid >> 5;           // row-strip id 0..15 (16 rows each)
    const int lane   = tid & 31;
    const int laneLo = lane & 15;
    const bool hi    = lane >= 16;
    const int s      = wave;

    // ---- preload A fragments for this strip (f32 16x4 A layout):
    // lanes 0-15: row=16s+lane, K={4kk,4kk+1}; lanes 16-31: K={4kk+2,4kk+3} ----
    v2f afrag[NK];
    {
        const float* qr = &sQ[(16 * s + laneLo) * DIM] + (hi ? 2 : 0);
        #pragma unroll
        for (int kk = 0; kk < NK; ++kk)
            afrag[kk] = *(const v2f*)(qr + 4 * kk);
    }
    float sq[8];
    #pragma unroll
    for (int r = 0; r < 8; ++r) sq[r] = sSq[16 * s + r + (hi ? 8 : 0)];

    float laneMax[8], laneSum[8], laneTheta[8];
    #pragma unroll
    for (int r = 0; r < 8; ++r) { laneMax[r] = -3.0e38f; laneSum[r] = 0.f; laneTheta[r] = 0.f; }

    // ---- pass 1: sim tiles via WMMA -> theta partials + online softmax stats of x=sim*Sigma ----
    for (int t = 0; t < NTILE; ++t) {
        // B 4x16 layout mirrors A: lanes 0-15 hold col=16t+lane, K={0,1}; hi lanes K={2,3}
        const float* kr = &sK[(16 * t + laneLo) * DIM] + (hi ? 2 : 0);
        // batch all B fragments first (8x ds_load_2addr_b64, one dscnt wait),
        // then 16 back-to-back wmma. sched_barrier keeps the scheduler from
        // re-interleaving load/wait pairs into the wmma chain.
        v2f bfrag[NK];
        #pragma unroll
        for (int kk = 0; kk < NK; ++kk)
            bfrag[kk] = *(const v2f*)(kr + 4 * kk);
        SCHED_FENCE();
        v8f C = {};
        #pragma unroll
        for (int kk = 0; kk < NK; ++kk)
            C = __builtin_amdgcn_wmma_f32_16x16x4_f32(false, afrag[kk], false, bfrag[kk],
                                                      (short)0, C, false, false);
        SCHED_FENCE();
        const float w1v = W1[16 * t + laneLo];
        const float skv = sSk[16 * t + laneLo];
        #pragma unroll
        for (int r = 0; r < 8; ++r) {
            const float simv = C[r];
            const int   iloc = r + (hi ? 8 : 0);
            const bool  diag = (t == s) && (laneLo == iloc);
            laneTheta[r] += diag ? 0.f : simv * w1v;
            const float x    = simv * sq[r] * skv;
            const float mOld = laneMax[r];
            const float mNew = fmaxf(mOld, x);
            laneSum[r] = laneSum[r] * __expf(mOld - mNew) + __expf(x - mNew);
            laneMax[r] = mNew;
        }
    }

    // ---- reduce row stats across the 16 lanes of each half (C-layout halves) ----
    #pragma unroll
    for (int r = 0; r < 8; ++r) {
        float m = laneMax[r], sm = laneSum[r], th = laneTheta[r];
        #pragma unroll
        for (int off = 1; off < 16; off <<= 1) {
            const float mo = __shfl_xor(m,  off, 32);
            const float so = __shfl_xor(sm, off, 32);
            const float to = __shfl_xor(th, off, 32);
            const float mn = fmaxf(m, mo);
            sm = sm * __expf(m - mn) + so * __expf(mo - mn);
            m  = mn;
            th += to;
        }
        if (laneLo == 0) {
            const int row = 16 * s + r + (hi ? 8 : 0);
            sRowMax[row] = m; sRowSum[row] = sm; sTheta[row] = th;
        }
    }
    __syncthreads();

    // ---- tiny MLP: hdn = W2a @ theta ; LeakyReLU(0.1) ; theta_scalar = W2b · hdn ----
    if (tid < LEN) {
        const float* wrow = &W2a[(size_t)tid * LEN];
        float acc = 0.f;
        #pragma unroll 8
        for (int i = 0; i < LEN; ++i) acc += wrow[i] * sTheta[i];
        acc = (acc >= 0.f) ? acc : 0.1f * acc;
        sRed[tid] = acc * W2b[tid];
    }
    __syncthreads();
    for (int stride = LEN / 2; stride > 0; stride >>= 1) {
        if (tid < stride) sRed[tid] += sRed[tid + stride];
        __syncthreads();
    }
    const float thetaS = sRed[0];

    // ---- pass 2: recompute sim via WMMA, apply Sigma, softmax + threshold mask, store ----
    float rM[8], rS[8];
    #pragma unroll
    for (int r = 0; r < 8; ++r) {
        const int row = 16 * s + r + (hi ? 8 : 0);
        rM[r] = sRowMax[row];
        rS[r] = 1.0f / sRowSum[row];
    }
    float* outg = out + (size_t)g * LEN * LEN;
    for (int t = 0; t < NTILE; ++t) {
        const float* kr = &sK[(16 * t + laneLo) * DIM] + (hi ? 2 : 0);
        v2f bfrag[NK];
        #pragma unroll
        for (int kk = 0; kk < NK; ++kk)
            bfrag[kk] = *(const v2f*)(kr + 4 * kk);
        SCHED_FENCE();
        v8f C = {};
        #pragma unroll
        for (int kk = 0; kk < NK; ++kk)
            C = __builtin_amdgcn_wmma_f32_16x16x4_f32(false, afrag[kk], false, bfrag[kk],
                                                      (short)0, C, false, false);
        SCHED_FENCE();
        const float skv = sSk[16 * t + laneLo];
        const int   col = 16 * t + laneLo;
        #pragma unroll
        for (int r = 0; r < 8; ++r) {
            const float x = C[r] * sq[r] * skv;
            float p = __expf(x - rM[r]) * rS[r];
            p = (x > thetaS) ? p : 0.f;
            const int row = 16 * s + r + (hi ? 8 : 0);
            outg[row * LEN + col] = p;
        }
    }
}

extern "C" void kernel_launch(void* const* d_in, const int* in_sizes, int n_in,
                              void* d_out, int out_size, void* d_ws, size_t ws_size,
                              hipStream_t stream) {
    const float* q   = (const float*)d_in[0];
    const float* k   = (const float*)d_in[1];
    const float* Wq  = (const float*)d_in[2];
    const float* bq  = (const float*)d_in[3];
    const float* Wk  = (const float*)d_in[4];
    const float* bk  = (const float*)d_in[5];
    const float* W1  = (const float*)d_in[6];
    const float* W2a = (const float*)d_in[7];
    const float* W2b = (const float*)d_in[8];
    float* out = (float*)d_out;

    const int groups = in_sizes[0] / (LEN * DIM);       // b*h*B = 512
    const size_t smem = (size_t)(2 * LEN * DIM + 6 * LEN) * sizeof(float); // ~137 KB

    // allow >64KB dynamic LDS (gfx1250 WGP has 320KB)
    (void)hipFuncSetAttribute((const void*)fused_sigma_attn,
                              hipFuncAttributeMaxDynamicSharedMemorySize, (int)smem);

    hipLaunchKernelGGL(fused_sigma_attn, dim3(groups), dim3(THREADS), smem, stream,
                       q, k, Wq, bq, Wk, bk, W1, W2a, W2b, out);
}